// VariationalAutoEncoder_14645838479504
// MI455X (gfx1250) — compile-verified
//
#include <hip/hip_runtime.h>
#include <hip/hip_bf16.h>
#include <stdint.h>

// Problem constants (fixed by the reference)
#define NBINS  32
#define VOX    (NBINS * NBINS * NBINS)   // 32768 voxels
#define SSAMP  512                       // samples per voxel
#define KSEL   (VOX / 2)                 // 16384 selected voxels
#define FPV    (SSAMP * 3)               // 1536 floats per voxel tile
#define V4PV   (FPV / 4)                 // 384 float4 per voxel tile
#define BYTES_PER_TILE (FPV * 4)         // 6144 bytes

#define MAIN_BLOCK 128
#define MAIN_GRID  4096
#define ROWS 3                           // float4s per thread per tile (384/128)

typedef float f32x4 __attribute__((ext_vector_type(4)));

// s_wait_asynccnt <= N  (CDNA5 split counter; ENDPGM does an implicit wait-idle)
#define WAIT_ASYNC(N) asm volatile("s_wait_asynccnt " #N ::: "memory")

// ---------------------------------------------------------------------------
// Prep: build slot->voxel compaction (prefix scan over prob_target>0),
// precompute per-slot {sigma[3], vox, mu[3]} and copy probs into out tail.
// ---------------------------------------------------------------------------
__global__ __launch_bounds__(1024)
void vae_prep(const float* __restrict__ prob_target,
              const float* __restrict__ enc_b,
              const float* __restrict__ vcent,
              float* __restrict__ params,      // [KSEL][8]
              float* __restrict__ probs_out)   // [VOX]
{
    __shared__ int part[1024];
    const int tid  = threadIdx.x;
    const int EPT  = VOX / 1024;               // 32 elements per thread
    const int base = tid * EPT;

    // Pad-with-voxel-0 default (jnp.nonzero(size=K) pads with index 0)
    {
        float s0 = enc_b[4 * VOX + 0], s1 = enc_b[4 * VOX + 1], s2 = enc_b[4 * VOX + 2];
        float m0 = enc_b[VOX + 0] + vcent[0];
        float m1 = enc_b[VOX + 1] + vcent[1];
        float m2 = enc_b[VOX + 2] + vcent[2];
        for (int k = tid; k < KSEL; k += 1024) {
            float* p = params + k * 8;
            p[0] = s0; p[1] = s1; p[2] = s2; p[3] = __int_as_float(0);
            p[4] = m0; p[5] = m1; p[6] = m2; p[7] = 0.0f;
        }
    }
    __syncthreads();

    int cnt = 0;
    for (int e = 0; e < EPT; ++e) cnt += (prob_target[base + e] > 0.0f) ? 1 : 0;
    part[tid] = cnt;
    __syncthreads();

    // Hillis-Steele inclusive scan over 1024 partials
    for (int off = 1; off < 1024; off <<= 1) {
        int v = (tid >= off) ? part[tid - off] : 0;
        __syncthreads();
        part[tid] += v;
        __syncthreads();
    }
    int slot = part[tid] - cnt;                // exclusive prefix

    for (int e = 0; e < EPT; ++e) {
        const int i = base + e;
        probs_out[i] = enc_b[i];               // probs = enc_b[:V] (inp is exactly 0)
        if (prob_target[i] > 0.0f && slot < KSEL) {
            float* p = params + slot * 8;
            p[0] = enc_b[4 * VOX + 3 * i + 0];                      // sigma
            p[1] = enc_b[4 * VOX + 3 * i + 1];
            p[2] = enc_b[4 * VOX + 3 * i + 2];
            p[3] = __int_as_float(i);                               // voxel index
            p[4] = enc_b[VOX + 3 * i + 0] + vcent[3 * i + 0];       // mu
            p[5] = enc_b[VOX + 3 * i + 1] + vcent[3 * i + 1];
            p[6] = enc_b[VOX + 3 * i + 2] + vcent[3 * i + 2];
            p[7] = 0.0f;
            ++slot;
        }
    }
}

// ---------------------------------------------------------------------------
// Issue one tile (6144 B) of eps into LDS via async global->LDS b128 copies.
// INST_OFFSET applies to BOTH the LDS and global address (ISA 15.18.3), so a
// single (lds_addr, global_offset) pair covers all three rows per lane.
// ---------------------------------------------------------------------------
__device__ __forceinline__
void async_copy_tile(uint32_t lds_addr, const float* eps_base, uint32_t glob_off)
{
    asm volatile(
        "global_load_async_to_lds_b128 %0, %1, %2 offset:0\n\t"
        "global_load_async_to_lds_b128 %0, %1, %2 offset:2048\n\t"
        "global_load_async_to_lds_b128 %0, %1, %2 offset:4096"
        :
        : "v"(lds_addr), "v"(glob_off), "s"(eps_base)
        : "memory");
}

// ---------------------------------------------------------------------------
// Main streaming kernel: double-buffered async LDS staging of eps tiles,
// FMA with per-voxel coefficients (component index rotates mod 3), NT stores.
// Each lane only reads the LDS slots its own async ops wrote -> no barriers.
// ---------------------------------------------------------------------------
__global__ __launch_bounds__(MAIN_BLOCK)
void vae_main(const float* __restrict__ eps,
              const float* __restrict__ params,
              f32x4* __restrict__ out,
              int nslots)
{
    __shared__ __align__(16) f32x4 buf[2][V4PV];    // 2 x 6144 B
    const int t = threadIdx.x;
    const uint32_t lds0 = (uint32_t)(uintptr_t)(&buf[0][0]) + (uint32_t)t * 16u;

    int slot = blockIdx.x;
    const int stride = gridDim.x;
    if (slot >= nslots) return;

    // Prime the pipeline: tile(slot) -> buf[0]
    {
        int vox = __float_as_int(params[slot * 8 + 3]);
        async_copy_tile(lds0, eps, (uint32_t)vox * BYTES_PER_TILE + (uint32_t)t * 16u);
    }

    int pb = 0;
    for (; slot < nslots; slot += stride) {
        const float* p8 = params + slot * 8;
        const float sg0 = p8[0], sg1 = p8[1], sg2 = p8[2];
        const float mu0 = p8[4], mu1 = p8[5], mu2 = p8[6];

        const int nslot = slot + stride;
        if (nslot < nslots) {
            int nvox = __float_as_int(params[nslot * 8 + 3]);
            async_copy_tile(lds0 + (uint32_t)(pb ^ 1) * (uint32_t)BYTES_PER_TILE,
                            eps, (uint32_t)nvox * BYTES_PER_TILE + (uint32_t)t * 16u);
            WAIT_ASYNC(3);      // 6 outstanding -> wait until only the new 3 remain
        } else {
            WAIT_ASYNC(0);      // drain the last tile
        }

        f32x4* obase = out + (size_t)slot * V4PV;
#pragma unroll
        for (int r = 0; r < ROWS; ++r) {
            const int j = t + r * MAIN_BLOCK;          // float4 index within tile
            const f32x4 e = buf[pb][j];                // ds_load_b128
            // float offset 4j -> components (j+k) mod 3
            const int p = j % 3;
            const float a0 = (p == 0) ? sg0 : ((p == 1) ? sg1 : sg2);
            const float a1 = (p == 0) ? sg1 : ((p == 1) ? sg2 : sg0);
            const float a2 = (p == 0) ? sg2 : ((p == 1) ? sg0 : sg1);
            const float b0 = (p == 0) ? mu0 : ((p == 1) ? mu1 : mu2);
            const float b1 = (p == 0) ? mu1 : ((p == 1) ? mu2 : mu0);
            const float b2 = (p == 0) ? mu2 : ((p == 1) ? mu0 : mu1);
            f32x4 r4;
            r4.x = fmaf(e.x, a0, b0);
            r4.y = fmaf(e.y, a1, b1);
            r4.z = fmaf(e.z, a2, b2);
            r4.w = fmaf(e.w, a0, b0);
            __builtin_nontemporal_store(r4, obase + j); // NT: no reuse, spare L2
        }
        pb ^= 1;
    }
}

// ---------------------------------------------------------------------------
// Host launch.
// Inputs: 0=x, 1=x_target, 2=prob_target, 3=eps, 4=enc_w, 5=enc_b, 6=voxel_centers
// Output: [ out (KSEL*FPV floats) | probs (VOX floats) ]
// ---------------------------------------------------------------------------
extern "C" void kernel_launch(void* const* d_in, const int* in_sizes, int n_in,
                              void* d_out, int out_size, void* d_ws, size_t ws_size,
                              hipStream_t stream)
{
    (void)in_sizes; (void)n_in; (void)out_size; (void)ws_size;
    const float* prob_target = (const float*)d_in[2];
    const float* eps         = (const float*)d_in[3];
    const float* enc_b       = (const float*)d_in[5];
    const float* vcent       = (const float*)d_in[6];

    float* out_f     = (float*)d_out;
    float* probs_out = out_f + (size_t)KSEL * FPV;
    float* params    = (float*)d_ws;               // KSEL*8 floats = 512 KB

    vae_prep<<<1, 1024, 0, stream>>>(prob_target, enc_b, vcent, params, probs_out);
    vae_main<<<MAIN_GRID, MAIN_BLOCK, 0, stream>>>(eps, params, (f32x4*)out_f, KSEL);
}